// CustomApplyTimeChannel_39960375722136
// MI455X (gfx1250) — compile-verified
//
#include <hip/hip_runtime.h>

typedef __attribute__((ext_vector_type(2))) float v2f;
typedef __attribute__((ext_vector_type(8))) float v8f;
typedef __attribute__((ext_vector_type(4))) int   v4i;

#define AS_GLOBAL __attribute__((address_space(1)))
#define AS_LDS    __attribute__((address_space(3)))

namespace {
constexpr int kBatch  = 128;
constexpr int kNB     = 14;
constexpr int kT      = 1096;
constexpr int kL      = 24;
constexpr int kFrame  = kT + kL - 1;                 // 1119
constexpr int kOutLen = (kNB - 1) * kT + kFrame;     // 15367
constexpr int kWaves  = 5;                           // ceil(ceil(1119/16)/16) tiles
constexpr int kThreads = kWaves * 32;                // 160 (wave32)
constexpr int kXWin   = 32 + 16 * 80;                // staged x window incl. halos = 1312
}  // namespace

// Zero only the positions that receive atomic adds: 15 seams of 23 samples per batch
// (frame heads t in [0,23) for n=0..13, plus the final frame tail which is seam 14).
__global__ void seam_zero_kernel(float* __restrict__ out) {
  const int idx = blockIdx.x * blockDim.x + threadIdx.x;
  const int total = kBatch * 15 * 23;
  if (idx >= total) return;
  const int b   = idx / (15 * 23);
  const int r   = idx % (15 * 23);
  const int seg = r / 23;
  const int q   = r % 23;
  out[(size_t)b * kOutLen + (size_t)seg * kT + q] = 0.0f;
}

__global__ __launch_bounds__(kThreads) void conv_oa_wmma(
    const float* __restrict__ x, const float* __restrict__ h,
    float* __restrict__ out) {
  __shared__ alignas(16) float sx[kXWin];
  __shared__ float sh[kL];

  const int bn = blockIdx.x;          // frame id: 0 .. B*NB-1
  const int b  = bn / kNB;
  const int n  = bn % kNB;
  const float* xp = x + (size_t)bn * kT;
  const float* hp = h + (size_t)bn * kL;

  // ---- Stage x once per frame into LDS: sx[32 + i] = x[i], zero halos. ----
#if __has_builtin(__builtin_amdgcn_global_load_async_to_lds_b128)
  // Zero the halos with plain DS stores (head 32 floats, tail 184 floats).
  for (int i = threadIdx.x; i < 32; i += kThreads) sx[i] = 0.0f;
  for (int i = threadIdx.x; i < kXWin - (32 + kT); i += kThreads)
    sx[32 + kT + i] = 0.0f;
  // Bulk body via CDNA5 async DMA to LDS (B128, both sides 16B aligned;
  // 1096 floats = 274 x b128, two sweeps of the 160-thread block).
  for (int i = 4 * (int)threadIdx.x; i < kT; i += 4 * kThreads) {
    __builtin_amdgcn_global_load_async_to_lds_b128(
        (AS_GLOBAL v4i*)(xp + i), (AS_LDS v4i*)(sx + 32 + i),
        /*offset=*/0, /*cpol=*/0);
  }
#if __has_builtin(__builtin_amdgcn_s_wait_asynccnt)
  __builtin_amdgcn_s_wait_asynccnt(0);
#else
  asm volatile("s_wait_asynccnt 0" ::: "memory");
#endif
#else
  for (int i = threadIdx.x; i < kXWin; i += kThreads) {
    const int xi = i - 32;
    sx[i] = (xi >= 0 && xi < kT) ? xp[xi] : 0.0f;
  }
#endif
  if (threadIdx.x < kL) sh[threadIdx.x] = hp[threadIdx.x];
  __syncthreads();

  const int lane = threadIdx.x & 31;
  const int wave = threadIdx.x >> 5;
  const int half = lane >> 4;   // 0: lanes 0-15, 1: lanes 16-31
  const int m    = lane & 15;

  // A fragments: three 16x16 Toeplitz tap matrices Tk[i,p] = h[16k + i - p],
  // each split along K into four 16x4 sub-matrices.
  // 16x4 fp32 A layout: lanes 0-15 hold K={0,1} (vgpr0,1), lanes 16-31 hold K={2,3}.
  v2f afrag[12];
#pragma unroll
  for (int k = 0; k < 3; ++k) {
#pragma unroll
    for (int k4 = 0; k4 < 4; ++k4) {
      const int p0 = 4 * k4 + 2 * half;     // global K column of .x
      const int h0 = 16 * k + m - p0;
      const int h1 = h0 - 1;
      v2f a;
      a.x = (h0 >= 0 && h0 < kL) ? sh[h0] : 0.0f;
      a.y = (h1 >= 0 && h1 < kL) ? sh[h1] : 0.0f;
      afrag[k * 4 + k4] = a;
    }
  }

  // One 16x16 output tile per wave: rows i = position within 16-chunk,
  // cols j = chunk index (jbase + j). Y[16c+i] = sum_k (Tk . X_{c-k})[i].
  const int jbase = wave * 16;
  const int j = m;                           // B/C/D column = lane % 16
  v8f acc = {};
#pragma unroll
  for (int k = 0; k < 3; ++k) {
    const int xbase = 32 + 16 * (jbase + j - k);   // LDS addr of x[16*(c-k)]
#pragma unroll
    for (int k4 = 0; k4 < 4; ++k4) {
      const int r = 4 * k4 + 2 * half;       // B row of .x within 16-row K dim
      v2f bf;
      bf.x = sx[xbase + r];
      bf.y = sx[xbase + r + 1];
      acc = __builtin_amdgcn_wmma_f32_16x16x4_f32(
          /*neg_a=*/false, afrag[k * 4 + k4], /*neg_b=*/false, bf,
          /*c_mod=*/(short)0, acc, /*reuse_a=*/false, /*reuse_b=*/false);
    }
  }

  // Write back. D vgpr v holds row (v + 8*half), column j = lane%16, so each
  // lane owns 8 CONSECUTIVE output samples starting at t0. The vast majority
  // of lanes (all of waves 1..3, most of waves 0/4) are fully interior and
  // exclusively owned -> one test, unconditional stores. Only the 23-sample
  // overlap seams and the frame tail take the per-element atomic path.
  const int t0 = 16 * (jbase + j) + 8 * half;
  float* outb = out + (size_t)b * kOutLen + (size_t)n * kT;
  if (t0 >= kL - 1 && t0 + 7 < kT) {
#pragma unroll
    for (int v = 0; v < 8; ++v) outb[t0 + v] = acc[v];
  } else {
#pragma unroll
    for (int v = 0; v < 8; ++v) {
      const int t = t0 + v;
      if (t < kFrame) {
        if (t < kL - 1 || t >= kT) {
          atomicAdd(&outb[t], acc[v]);       // overlap seams / single-owner edges
        } else {
          outb[t] = acc[v];                  // exclusively owned interior
        }
      }
    }
  }
}

extern "C" void kernel_launch(void* const* d_in, const int* in_sizes, int n_in,
                              void* d_out, int out_size, void* d_ws, size_t ws_size,
                              hipStream_t stream) {
  const float* x = (const float*)d_in[0];   // [128, 14, 1096] f32
  const float* h = (const float*)d_in[1];   // [128, 14, 24]   f32
  float* out = (float*)d_out;               // [128, 15367]    f32

  const int seam_total = kBatch * 15 * 23;
  seam_zero_kernel<<<(seam_total + 255) / 256, 256, 0, stream>>>(out);
  conv_oa_wmma<<<kBatch * kNB, kThreads, 0, stream>>>(x, h, out);
}